// RoIPool_29137058136180
// MI455X (gfx1250) — compile-verified
//
#include <hip/hip_runtime.h>
#include <math.h>

// RoIPool (max) for MI455X / gfx1250.
// input: (N=4, C=256, H=64, W=64) fp32, NCHW
// rois : (K=2048, 5) fp32  [batch, x1, y1, x2, y2] in image coords (scale 1/16)
// out  : (K, C, 7, 7) fp32
//
// Mapping: 1 block per ROI, 256 threads (8 wave32s), thread = channel.
//  - ROI geometry is block-uniform -> compiler puts it on the gfx1250 SALU
//    float pipe (s_mul_f32 / s_rndne_f32 / s_cvt_i32_f32), once per wave.
//  - Input slab (16.8 MB) is resident in the 192 MB L2; gathers stream along
//    w so successive iterations hit the same WGP$ line.
//  - Output (103 MB) is write-once -> non-temporal stores (th:TH_STORE_NT)
//    keep the streaming writes from churning L2 while per-thread runs of 49
//    consecutive dwords still fill whole 128B lines for HBM.

#define ROIP_PH 7
#define ROIP_PW 7
#define ROIP_SCALE 0.0625f
#define ROIP_C 256
#define ROIP_H 64
#define ROIP_W 64

__global__ __launch_bounds__(ROIP_C) void roipool_max_kernel(
    const float* __restrict__ input,
    const float* __restrict__ rois,
    float* __restrict__ out)
{
    const int k = blockIdx.x;   // ROI index
    const int c = threadIdx.x;  // channel

    // --- ROI params (uniform across block; scalarized by the compiler) ---
    const float rb  = rois[k * 5 + 0];
    const float rx1 = rois[k * 5 + 1];
    const float ry1 = rois[k * 5 + 2];
    const float rx2 = rois[k * 5 + 3];
    const float ry2 = rois[k * 5 + 4];

    const int n    = (int)rb;                       // astype(int32) truncation
    const int rs_w = (int)rintf(rx1 * ROIP_SCALE);  // s_rndne_f32 == jnp.round
    const int rs_h = (int)rintf(ry1 * ROIP_SCALE);
    const int re_w = (int)rintf(rx2 * ROIP_SCALE);
    const int re_h = (int)rintf(ry2 * ROIP_SCALE);

    const int roi_w = max(re_w - rs_w + 1, 1);
    const int roi_h = max(re_h - rs_h + 1, 1);
    const float bin_h = (float)roi_h / (float)ROIP_PH;
    const float bin_w = (float)roi_w / (float)ROIP_PW;

    const float* __restrict__ base =
        input + ((size_t)n * ROIP_C + (size_t)c) * (ROIP_H * ROIP_W);
    float* __restrict__ outp =
        out + ((size_t)k * ROIP_C + (size_t)c) * (ROIP_PH * ROIP_PW);

    // --- per-pw bin bounds (block-uniform) ---
    int wstart[ROIP_PW], wend[ROIP_PW];
#pragma unroll
    for (int pw = 0; pw < ROIP_PW; ++pw) {
        int ws = (int)floorf((float)pw * bin_w) + rs_w;
        int we = (int)ceilf((float)(pw + 1) * bin_w) + rs_w;
        wstart[pw] = min(max(ws, 0), ROIP_W);
        wend[pw]   = min(max(we, 0), ROIP_W);
    }

    // Warm the cache hierarchy for this channel's ROI region
    // (lowers to global_prefetch_b8 on gfx1250).
    {
        const int hp = min(max(rs_h, 0), ROIP_H - 1);
        const int wp = min(max(rs_w, 0), ROIP_W - 1);
        __builtin_prefetch(base + hp * ROIP_W + wp, 0, 3);
        const int hp2 = min(hp + 4, ROIP_H - 1);
        __builtin_prefetch(base + hp2 * ROIP_W + wp, 0, 3);
    }

    for (int ph = 0; ph < ROIP_PH; ++ph) {
        int hs = (int)floorf((float)ph * bin_h) + rs_h;
        int he = (int)ceilf((float)(ph + 1) * bin_h) + rs_h;
        hs = min(max(hs, 0), ROIP_H);
        he = min(max(he, 0), ROIP_H);

        for (int pw = 0; pw < ROIP_PW; ++pw) {
            const int ws  = wstart[pw];
            const int we  = wend[pw];
            const int len = we - ws;

            float m = -INFINITY;
            for (int h = hs; h < he; ++h) {
                const float* __restrict__ row = base + h * ROIP_W + ws;
                int w = 0;
                for (; w + 3 < len; w += 4) {
                    float a0 = row[w + 0];
                    float a1 = row[w + 1];
                    float a2 = row[w + 2];
                    float a3 = row[w + 3];
                    m = fmaxf(m, fmaxf(fmaxf(a0, a1), fmaxf(a2, a3)));
                }
                for (; w < len; ++w) m = fmaxf(m, row[w]);
            }
            // Empty bin -> 0.0 (matches reference's any(mask) fallback).
            const float r = (hs < he && len > 0) ? m : 0.0f;
            // Write-once output stream: non-temporal store (TH_STORE_NT).
            __builtin_nontemporal_store(r, &outp[ph * ROIP_PW + pw]);
        }
    }
}

extern "C" void kernel_launch(void* const* d_in, const int* in_sizes, int n_in,
                              void* d_out, int out_size, void* d_ws, size_t ws_size,
                              hipStream_t stream) {
    const float* input = (const float*)d_in[0];  // (N, C, H, W) fp32
    const float* rois  = (const float*)d_in[1];  // (K, 5) fp32
    float* out         = (float*)d_out;          // (K, C, 7, 7) fp32

    const int K = in_sizes[1] / 5;  // 2048

    roipool_max_kernel<<<dim3(K), dim3(ROIP_C), 0, stream>>>(input, rois, out);
}